// Attention_4148938408202
// MI455X (gfx1250) — compile-verified
//
#include <hip/hip_runtime.h>

typedef __attribute__((ext_vector_type(2))) float v2f;
typedef __attribute__((ext_vector_type(8))) float v8f;

#define BB      32
#define HQn     32
#define HKVn    8
#define Gn      4
#define Dn      128
#define Sn      4096
#define NSPLIT  4
#define CHUNK   (Sn / NSPLIT)   // 1024
#define NW      4               // waves per block (wave32)
#define M_INIT  (-1.0e30f)

// qk_scale * log2(e): scores computed directly in log2 domain
#define QCOEF   (0.08838834764831845f * 1.4426950408889634f)

__global__ __launch_bounds__(NW * 32, 1) void attn_partial_kernel(
    const float* __restrict__ xq, const float* __restrict__ xk,
    const float* __restrict__ xv, const float* __restrict__ kvbuf,
    const int* __restrict__ cur_sel, const int* __restrict__ table,
    const int* __restrict__ seqlen,
    float* __restrict__ ws_m, float* __restrict__ ws_l,
    float* __restrict__ ws_acc)
{
  __shared__ float lm[NW][Gn];
  __shared__ float ll[NW][Gn];
  __shared__ float lacc[NW][Gn][Dn];

  const int bh    = blockIdx.x;
  const int split = blockIdx.y;
  const int b     = bh / HKVn;
  const int h     = bh % HKVn;
  const int wave  = threadIdx.x >> 5;
  const int lane  = threadIdx.x & 31;
  const int half  = lane >> 4;
  const int ln    = lane & 15;

  const int len    = seqlen[b];
  const int csel   = cur_sel[b];
  const int cstart = split * CHUNK;
  const int vend   = min(len, cstart + CHUNK);
  const int nvalid = vend - cstart;
  const int ntiles = (nvalid > 0) ? ((nvalid + 15) >> 4) : 0;

  // ---- Q as resident WMMA B-operands (d-permuted: low lanes d 0..63, high d 64..127)
  v2f bq[32];
  {
    const float* qp = xq + (size_t)(b * HQn + h * Gn + ln) * Dn + 64 * half;
    #pragma unroll
    for (int c = 0; c < 32; ++c) {
      float x0 = 0.f, x1 = 0.f;
      if (ln < Gn) { x0 = qp[2 * c] * QCOEF; x1 = qp[2 * c + 1] * QCOEF; }
      bq[c][0] = x0; bq[c][1] = x1;
    }
  }

  float m = M_INIT, l = 0.f;
  v8f acc[8];
  #pragma unroll
  for (int dt = 0; dt < 8; ++dt)
    #pragma unroll
    for (int r = 0; r < 8; ++r) acc[dt][r] = 0.f;

  const float* xkRow = xk + (size_t)(b * HKVn + h) * Dn;
  const float* xvRow = xv + (size_t)(b * HKVn + h) * Dn;

  for (int t = wave; t < ntiles; t += NW) {
    const int s0 = cstart + (t << 4);

    // table row for this lane's tile row (clamped; invalid rows are masked later)
    const int sK  = min(s0 + ln, vend - 1);
    const int row = table[(size_t)b * Sn + sK];
    const float* kbase = (row == csel) ? xkRow
                        : (kvbuf + ((size_t)row * (2 * HKVn) + h) * Dn);
    const float* kptr = kbase + 64 * half;

    // V row pointers for tile rows 8*half + j  (j = 0..7)
    const float* vbase[8];
    #pragma unroll
    for (int j = 0; j < 8; ++j) {
      int r       = 8 * half + j;
      int srcRow  = __shfl(row, r, 32);   // lane r holds tile-row r's table entry
      vbase[j] = (srcRow == csel) ? xvRow
               : (kvbuf + ((size_t)srcRow * (2 * HKVn) + HKVn + h) * Dn);
    }

    // ---- QK^T: 32 chunks of K=4 over D, scores in log2 domain
    v8f C;
    #pragma unroll
    for (int r = 0; r < 8; ++r) C[r] = 0.f;

    #pragma unroll
    for (int bt = 0; bt < 2; ++bt) {
      float4 ka[8];
      #pragma unroll
      for (int j = 0; j < 8; ++j)
        ka[j] = *(const float4*)(kptr + (bt * 8 + j) * 4);
      #pragma unroll
      for (int j = 0; j < 8; ++j) {
        int c0 = (bt * 8 + j) * 2;
        v2f a0; a0[0] = ka[j].x; a0[1] = ka[j].y;
        v2f a1; a1[0] = ka[j].z; a1[1] = ka[j].w;
        C = __builtin_amdgcn_wmma_f32_16x16x4_f32(false, a0, false, bq[c0],
                                                  (short)0, C, false, false);
        C = __builtin_amdgcn_wmma_f32_16x16x4_f32(false, a1, false, bq[c0 + 1],
                                                  (short)0, C, false, false);
      }
    }

    // ---- mask + online softmax (per g column; cross-half via shfl_xor 16)
    #pragma unroll
    for (int r = 0; r < 8; ++r) {
      int s = s0 + r + 8 * half;
      C[r] = (s < vend) ? C[r] : M_INIT;
    }
    float tm = C[0];
    #pragma unroll
    for (int r = 1; r < 8; ++r) tm = fmaxf(tm, C[r]);
    tm = fmaxf(tm, __shfl_xor(tm, 16, 32));
    float mn = fmaxf(m, tm);
    float f  = __builtin_amdgcn_exp2f(m - mn);
    float ts = 0.f;
    #pragma unroll
    for (int r = 0; r < 8; ++r) {
      float p = __builtin_amdgcn_exp2f(C[r] - mn);
      C[r] = p;                 // C VGPRs are now the PV B-operands directly
      ts += p;
    }
    ts += __shfl_xor(ts, 16, 32);
    l = l * f + ts;
    m = mn;
    #pragma unroll
    for (int dt = 0; dt < 8; ++dt)
      #pragma unroll
      for (int r = 0; r < 8; ++r) acc[dt][r] *= f;

    // ---- PV: O^T tiles, seq permuted per chunk {2c,2c+1,8+2c,9+2c}
    #pragma unroll
    for (int dt = 0; dt < 8; ++dt) {
      float va[8];
      #pragma unroll
      for (int c = 0; c < 4; ++c) {
        va[2 * c]     = vbase[2 * c][dt * 16 + ln];
        va[2 * c + 1] = vbase[2 * c + 1][dt * 16 + ln];
      }
      #pragma unroll
      for (int c = 0; c < 4; ++c) {
        v2f a;  a[0] = va[2 * c]; a[1] = va[2 * c + 1];
        v2f bp; bp[0] = C[2 * c]; bp[1] = C[2 * c + 1];
        acc[dt] = __builtin_amdgcn_wmma_f32_16x16x4_f32(false, a, false, bp,
                                                        (short)0, acc[dt],
                                                        false, false);
      }
    }
  }

  // ---- per-wave partial -> LDS
  if (ln < Gn) {
    #pragma unroll
    for (int dt = 0; dt < 8; ++dt)
      #pragma unroll
      for (int r = 0; r < 8; ++r)
        lacc[wave][ln][dt * 16 + r + 8 * half] = acc[dt][r];
    if (half == 0) { lm[wave][ln] = m; ll[wave][ln] = l; }
  }
  __syncthreads();

  // ---- merge NW wave-partials -> one partial per (bh, split)
  const int tid = threadIdx.x;           // 0..127 == d
  float mg[Gn];
  #pragma unroll
  for (int g = 0; g < Gn; ++g) {
    float mm = lm[0][g];
    #pragma unroll
    for (int w = 1; w < NW; ++w) mm = fmaxf(mm, lm[w][g]);
    mg[g] = mm;
  }
  #pragma unroll
  for (int g = 0; g < Gn; ++g) {
    float s = 0.f;
    #pragma unroll
    for (int w = 0; w < NW; ++w)
      s += __builtin_amdgcn_exp2f(lm[w][g] - mg[g]) * lacc[w][g][tid];
    ws_acc[(((size_t)bh * NSPLIT + split) * Gn + g) * Dn + tid] = s;
  }
  if (tid < Gn) {
    float s = 0.f;
    #pragma unroll
    for (int w = 0; w < NW; ++w)
      s += __builtin_amdgcn_exp2f(lm[w][tid] - mg[tid]) * ll[w][tid];
    ws_l[((size_t)bh * NSPLIT + split) * Gn + tid] = s;
    ws_m[((size_t)bh * NSPLIT + split) * Gn + tid] = mg[tid];
  }
}

__global__ __launch_bounds__(Dn, 1) void attn_combine_kernel(
    const float* __restrict__ ws_m, const float* __restrict__ ws_l,
    const float* __restrict__ ws_acc, float* __restrict__ out)
{
  const int bh = blockIdx.x;
  const int b  = bh / HKVn;
  const int h  = bh % HKVn;
  const int d  = threadIdx.x;
  #pragma unroll
  for (int g = 0; g < Gn; ++g) {
    float mm = M_INIT;
    #pragma unroll
    for (int s = 0; s < NSPLIT; ++s)
      mm = fmaxf(mm, ws_m[((size_t)bh * NSPLIT + s) * Gn + g]);
    float num = 0.f, den = 0.f;
    #pragma unroll
    for (int s = 0; s < NSPLIT; ++s) {
      float f = __builtin_amdgcn_exp2f(ws_m[((size_t)bh * NSPLIT + s) * Gn + g] - mm);
      den += f * ws_l[((size_t)bh * NSPLIT + s) * Gn + g];
      num += f * ws_acc[(((size_t)bh * NSPLIT + s) * Gn + g) * Dn + d];
    }
    out[(size_t)(b * HQn + h * Gn + g) * Dn + d] = num / den;
  }
}

extern "C" void kernel_launch(void* const* d_in, const int* in_sizes, int n_in,
                              void* d_out, int out_size, void* d_ws, size_t ws_size,
                              hipStream_t stream)
{
  (void)in_sizes; (void)n_in; (void)out_size; (void)ws_size;
  const float* xq     = (const float*)d_in[0];
  const float* xk     = (const float*)d_in[1];
  const float* xv     = (const float*)d_in[2];
  const float* kvbuf  = (const float*)d_in[3];
  const int*   csel   = (const int*)d_in[4];
  const int*   table  = (const int*)d_in[5];
  const int*   seqlen = (const int*)d_in[6];
  float* out = (float*)d_out;

  float* ws_m   = (float*)d_ws;                                  // 256*4*4
  float* ws_l   = ws_m + (size_t)BB * HKVn * NSPLIT * Gn;        // 256*4*4
  float* ws_acc = ws_l + (size_t)BB * HKVn * NSPLIT * Gn;        // 256*4*4*128

  dim3 grid1(BB * HKVn, NSPLIT);
  attn_partial_kernel<<<grid1, NW * 32, 0, stream>>>(
      xq, xk, xv, kvbuf, csel, table, seqlen, ws_m, ws_l, ws_acc);
  attn_combine_kernel<<<BB * HKVn, Dn, 0, stream>>>(ws_m, ws_l, ws_acc, out);
}